// Loss_61675730370853
// MI455X (gfx1250) — compile-verified
//
#include <hip/hip_runtime.h>
#include <hip/hip_bf16.h>

#define BATCH   16
#define NA      32768
#define NG      32
#define TPB     256
#define CHUNKS  16                    // blocks per sample
#define APT     (NA / (CHUNKS * TPB)) // anchors per thread = 8

typedef float v2f __attribute__((ext_vector_type(2)));
typedef float v8f __attribute__((ext_vector_type(8)));

// ---------------------------------------------------------------------------
// Kernel 0: zero the per-(b,g) argmax keys
// ---------------------------------------------------------------------------
__global__ void init_best_kernel(unsigned long long* __restrict__ best) {
    int t = threadIdx.x;
    if (t < BATCH * NG) best[t] = 0ULL;
}

__device__ __forceinline__ float iou_f(float ax0, float ay0, float ax1, float ay1, float aar,
                                       float gx0, float gy0, float gx1, float gy1, float gar) {
    float lx = fmaxf(ax0, gx0), ly = fmaxf(ay0, gy0);
    float rx = fminf(ax1, gx1), ry = fminf(ay1, gy1);
    float iw = fmaxf(rx - lx, 0.0f), ih = fmaxf(ry - ly, 0.0f);
    float inter = iw * ih;
    return inter / (aar + gar - inter);
}

// ---------------------------------------------------------------------------
// Kernel 1: per-(sample, gt) argmax over all anchors (packed u64 max keys)
// ---------------------------------------------------------------------------
__global__ void argmax_kernel(const float* __restrict__ anchors,
                              const float* __restrict__ gt,
                              const int*   __restrict__ nobj,
                              unsigned long long* __restrict__ best) {
    const int b     = blockIdx.x / CHUNKS;
    const int chunk = blockIdx.x % CHUNKS;
    const int t     = threadIdx.x;

    __shared__ float              gsh[NG * 5];   // x0,y0,x1,y1,area
    __shared__ unsigned long long bsh[NG];

    if (t < NG) {
        const float* g4 = gt + ((size_t)b * NG + t) * 4;
        float x = g4[0], y = g4[1], w = g4[2], h = g4[3];
        gsh[t * 5 + 0] = x;     gsh[t * 5 + 1] = y;
        gsh[t * 5 + 2] = x + w; gsh[t * 5 + 3] = y + h;
        gsh[t * 5 + 4] = w * h;
        bsh[t] = 0ULL;
    }
    __syncthreads();

    const int base = chunk * (TPB * APT) + t;
    float ax0[APT], ay0[APT], ax1[APT], ay1[APT], aar[APT];
#pragma unroll
    for (int i = 0; i < APT; ++i) {
        const float* a4 = anchors + (size_t)(base + i * TPB) * 4;
        ax0[i] = a4[0]; ay0[i] = a4[1]; ax1[i] = a4[2]; ay1[i] = a4[3];
        aar[i] = (ax1[i] - ax0[i]) * (ay1[i] - ay0[i]);
    }

    const int n = nobj[b];
    for (int g = 0; g < n; ++g) {
        float gx0 = gsh[g * 5 + 0], gy0 = gsh[g * 5 + 1];
        float gx1 = gsh[g * 5 + 2], gy1 = gsh[g * 5 + 3];
        float gar = gsh[g * 5 + 4];
        unsigned long long kmax = 0ULL;
#pragma unroll
        for (int i = 0; i < APT; ++i) {
            float iou = iou_f(ax0[i], ay0[i], ax1[i], ay1[i], aar[i],
                              gx0, gy0, gx1, gy1, gar);
            unsigned long long key =
                ((unsigned long long)__float_as_uint(iou) << 32) |
                (unsigned long long)(0xFFFFFFFFu - (unsigned)(base + i * TPB));
            kmax = (key > kmax) ? key : kmax;
        }
        atomicMax(&bsh[g], kmax);    // LDS ds_max_u64
    }
    __syncthreads();
    if (t < NG) atomicMax(&best[b * NG + t], bsh[t]);  // global_atomic_max_u64
}

// ---------------------------------------------------------------------------
// Kernel 2: focal + coord partial sums -> partials[q][chunk(16)][sample(16)]
// ---------------------------------------------------------------------------
__global__ void loss_kernel(const float* __restrict__ boxes,
                            const float* __restrict__ classes,
                            const float* __restrict__ anchors,
                            const float* __restrict__ gt,
                            const int*   __restrict__ nobj,
                            const unsigned long long* __restrict__ best,
                            float* __restrict__ partials) {
    const int b     = blockIdx.x / CHUNKS;
    const int chunk = blockIdx.x % CHUNKS;
    const int t     = threadIdx.x;

    __shared__ float        gsh[NG * 5];
    __shared__ unsigned int bestA[NG];
    __shared__ float        red[8 * 3];

    if (t < NG) {
        const float* g4 = gt + ((size_t)b * NG + t) * 4;
        float x = g4[0], y = g4[1], w = g4[2], h = g4[3];
        gsh[t * 5 + 0] = x;     gsh[t * 5 + 1] = y;
        gsh[t * 5 + 2] = x + w; gsh[t * 5 + 3] = y + h;
        gsh[t * 5 + 4] = w * h;
        unsigned long long k = best[b * NG + t];
        bestA[t] = 0xFFFFFFFFu - (unsigned)(k & 0xFFFFFFFFull);
    }
    __syncthreads();

    const int n = nobj[b];
    float clsAcc = 0.0f, crdAcc = 0.0f, nprAcc = 0.0f;
    const int base = chunk * (TPB * APT) + t;

    for (int i = 0; i < APT; ++i) {
        const int a = base + i * TPB;
        if (i + 1 < APT)
            __builtin_prefetch(boxes + ((size_t)b * NA + (a + TPB)) * 4, 0, 0);

        const float* a4 = anchors + (size_t)a * 4;
        float ax0 = a4[0], ay0 = a4[1], ax1 = a4[2], ay1 = a4[3];
        float aar = (ax1 - ax0) * (ay1 - ay0);
        const float* p4 = boxes + ((size_t)b * NA + a) * 4;
        float bx0 = p4[0], bx1 = p4[1], bx2 = p4[2], bx3 = p4[3];

        bool posAny = false;
        for (int g = 0; g < n; ++g) {
            float gx0 = gsh[g * 5 + 0], gy0 = gsh[g * 5 + 1];
            float gx1 = gsh[g * 5 + 2], gy1 = gsh[g * 5 + 3];
            float gar = gsh[g * 5 + 4];
            float iou = iou_f(ax0, ay0, ax1, ay1, aar, gx0, gy0, gx1, gy1, gar);
            bool pos = (iou > 0.5f) | (bestA[g] == (unsigned)a);
            if (pos) {
                posAny = true;
                nprAcc += 1.0f;
                crdAcc += fabsf(bx0 - gx0) + fabsf(bx1 - gy0) +
                          fabsf(bx2 - gx1) + fabsf(bx3 - gy1);
            }
        }
        // focal loss, 2 classes, weight = {1, 5}, gamma = 2
        const float* c2 = classes + ((size_t)b * NA + a) * 2;
        float c0 = c2[0], c1 = c2[1];
        float m   = fmaxf(c0, c1);
        float lse = m + __logf(__expf(c0 - m) + __expf(c1 - m));
        float ct  = posAny ? c1 : c0;
        float w   = posAny ? 5.0f : 1.0f;
        float logt = ct - lse;
        float pt   = __expf(logt);
        float om   = 1.0f - pt;
        clsAcc += -w * om * om * logt;
    }

    // wave32 shuffle tree
#pragma unroll
    for (int off = 16; off > 0; off >>= 1) {
        clsAcc += __shfl_xor(clsAcc, off, 32);
        crdAcc += __shfl_xor(crdAcc, off, 32);
        nprAcc += __shfl_xor(nprAcc, off, 32);
    }
    const int wave = t >> 5, lane = t & 31;
    if (lane == 0) {
        red[wave * 3 + 0] = clsAcc;
        red[wave * 3 + 1] = crdAcc;
        red[wave * 3 + 2] = nprAcc;
    }
    __syncthreads();
    if (t == 0) {
        float c = 0.0f, d = 0.0f, np = 0.0f;
        for (int wv = 0; wv < 8; ++wv) {
            c  += red[wv * 3 + 0];
            d  += red[wv * 3 + 1];
            np += red[wv * 3 + 2];
        }
        partials[0 * (CHUNKS * BATCH) + chunk * BATCH + b] = c;
        partials[1 * (CHUNKS * BATCH) + chunk * BATCH + b] = d;
        partials[2 * (CHUNKS * BATCH) + chunk * BATCH + b] = np;
    }
}

// ---------------------------------------------------------------------------
// Kernel 3: finalize. Reduce 16x16 partials per quantity on the matrix pipe:
//   D = ones(16x4) x P(4x16) + C, chained 4x  (V_WMMA_F32_16X16X4_F32).
// One wave, EXEC all-ones (no divergence before the WMMAs).
// ---------------------------------------------------------------------------
__global__ void finalize_kernel(const float* __restrict__ partials,
                                float* __restrict__ out) {
    const int lane = threadIdx.x;          // 0..31
    const int col  = lane & 15;            // sample index
    const int r0   = (lane >> 4) * 2;      // rows {0,1} or {2,3} of each 4-slab

    v2f ones; ones.x = 1.0f; ones.y = 1.0f;
    float sums[3];
#pragma unroll
    for (int q = 0; q < 3; ++q) {
        v8f acc = {0.0f, 0.0f, 0.0f, 0.0f, 0.0f, 0.0f, 0.0f, 0.0f};
#pragma unroll
        for (int chunk = 0; chunk < 4; ++chunk) {
            v2f bv;
            bv.x = partials[q * 256 + (chunk * 4 + r0 + 0) * 16 + col];
            bv.y = partials[q * 256 + (chunk * 4 + r0 + 1) * 16 + col];
            acc = __builtin_amdgcn_wmma_f32_16x16x4_f32(
                false, ones, false, bv, (short)0, acc, false, false);
        }
        sums[q] = acc[0];   // column sum, replicated over rows
    }

    float clsb = sums[0] * (1.0f / (float)NA);   // mean over anchors
    float crdb = sums[1] / (sums[2] * 4.0f);     // coord sum / (n_pairs*4)
    float mask = (lane < 16) ? 1.0f : 0.0f;      // kill mirrored half-wave
    clsb *= mask; crdb *= mask;
#pragma unroll
    for (int off = 16; off > 0; off >>= 1) {
        clsb += __shfl_xor(clsb, off, 32);
        crdb += __shfl_xor(crdb, off, 32);
    }
    if (lane == 0) {
        float cl = clsb * (1.0f / (float)BATCH);
        float cr = crdb * (1.0f / (float)BATCH);
        out[0] = cl + cr;   // total_loss
        out[1] = cl;        // class_loss
        out[2] = cr;        // coord_loss
    }
}

// ---------------------------------------------------------------------------
extern "C" void kernel_launch(void* const* d_in, const int* in_sizes, int n_in,
                              void* d_out, int out_size, void* d_ws, size_t ws_size,
                              hipStream_t stream) {
    const float* boxes   = (const float*)d_in[0];   // [B,A,4]
    const float* classes = (const float*)d_in[1];   // [B,A,2]
    const float* anchors = (const float*)d_in[2];   // [A,4] xyxy
    const float* gt      = (const float*)d_in[3];   // [B,G,4] xywh
    const int*   nobj    = (const int*)d_in[4];     // [B]
    float*       out     = (float*)d_out;           // 3 floats

    unsigned long long* best = (unsigned long long*)d_ws;              // 512 * 8B
    float* partials = (float*)((char*)d_ws + BATCH * NG * sizeof(unsigned long long));
                                                                       // 3*16*16 floats

    init_best_kernel<<<1, 512, 0, stream>>>(best);
    argmax_kernel<<<BATCH * CHUNKS, TPB, 0, stream>>>(anchors, gt, nobj, best);
    loss_kernel<<<BATCH * CHUNKS, TPB, 0, stream>>>(boxes, classes, anchors, gt,
                                                    nobj, best, partials);
    finalize_kernel<<<1, 32, 0, stream>>>(partials, out);
}